// GraphMMCorrector_56057913147629
// MI455X (gfx1250) — compile-verified
//
#include <hip/hip_runtime.h>
#include <math.h>

// ---------------- problem constants ----------------
#define NN   50000     // nodes
#define BB   16        // batch
#define LL   128       // seq len
#define DD   64        // hidden
#define EEC  200000    // edges
#define ETT  100000    // traj edges
#define MLPH 128
#define TEMP 12.0f
#define NTILES_N (NN / 16)   // 3125 logits tiles

typedef __attribute__((ext_vector_type(2))) float v2f;
typedef __attribute__((ext_vector_type(8))) float v8f;

// =====================================================================
// f32 WMMA GEMM:  OUT[M,NC] = act( scale * ((X (+Xadd)) @ W + bias) )
// One wave computes one 16x16 tile via V_WMMA_F32_16X16X4_F32.
// A layout (ISA 7.12.2, 16x4 f32): lanes0-15 row=lane, regs={K=k0,k0+1};
//                                  lanes16-31 row=lane-16, regs={K=k0+2,k0+3}.
// B layout mirrored (4x16): same half split on K, col=lane%16.
// C/D layout: reg i -> row = i + (lane/16)*8, col = lane%16.
// Chunked: prefetch 16 k-steps of fragments (b64 loads), then 16 WMMAs.
// PACKEDB: W pre-packed in fragment order -> one contiguous v2f per lane.
// ARGMAX:  per-tile per-row (value, col) partials via xor-shuffle reduce.
// =====================================================================
template<int K, int NC, bool HASADD, bool PACKEDB, bool ARGMAX>
__global__ void gemm_wmma_kernel(const float* __restrict__ X,
                                 const float* __restrict__ Xadd,
                                 const float* __restrict__ W,
                                 const float* __restrict__ bias,
                                 float* __restrict__ OUT,
                                 int M, long long ldOut, float scale, int relu,
                                 float* __restrict__ PV, int* __restrict__ PI)
{
    constexpr int STEPS = K / 4;          // k-steps total (16 or 32)
    constexpr int CH = 16;                // k-steps per prefetch chunk
    constexpr int tilesN = NC >> 4;
    int wave = threadIdx.x >> 5;
    int lane = threadIdx.x & 31;
    int totalTiles = (M >> 4) * tilesN;
    int tile = blockIdx.x * (blockDim.x >> 5) + wave;
    if (tile >= totalTiles) return;       // wave-uniform exit
    int tm = tile / tilesN;
    int tn = tile - tm * tilesN;
    int half = lane >> 4;
    int r    = lane & 15;
    long long rowA = (long long)(tm * 16 + r);
    int colB = tn * 16 + r;
    const float* xrow = X + rowA * K;
    const float* arow = Xadd + rowA * K;  // only read when HASADD

    v8f acc = {0.f,0.f,0.f,0.f,0.f,0.f,0.f,0.f};
    #pragma unroll
    for (int c = 0; c < STEPS / CH; ++c) {
        v2f afr[CH], bfr[CH];
        // ---- burst-load all fragments for this chunk ----
        #pragma unroll
        for (int s = 0; s < CH; ++s) {
            int ka = c * (CH * 4) + s * 4 + half * 2;
            v2f a = *(const v2f*)(xrow + ka);
            if (HASADD) {
                v2f ad = *(const v2f*)(arow + ka);
                a.x += ad.x; a.y += ad.y;
            }
            afr[s] = a;
            if (PACKEDB) {
                bfr[s] = ((const v2f*)W)[((long long)tn * STEPS + c * CH + s) * 32 + lane];
            } else {
                v2f b;
                b.x = W[(long long)ka * NC + colB];
                b.y = W[(long long)(ka + 1) * NC + colB];
                bfr[s] = b;
            }
        }
        // ---- WMMA accumulate chain ----
        #pragma unroll
        for (int s = 0; s < CH; ++s) {
            acc = __builtin_amdgcn_wmma_f32_16x16x4_f32(
                false, afr[s], false, bfr[s], (short)0, acc, false, false);
        }
    }
    // ---- epilogue: bias/scale/relu + store (+ partial argmax) ----
    float vals[8];
    #pragma unroll
    for (int i = 0; i < 8; ++i) {
        float v = acc[i];
        if (bias) v += bias[colB];
        v *= scale;
        if (relu) v = fmaxf(v, 0.f);
        vals[i] = v;
        int orow = tm * 16 + i + half * 8;
        OUT[(long long)orow * ldOut + colB] = v;
    }
    if (ARGMAX) {
        #pragma unroll
        for (int i = 0; i < 8; ++i) {
            float v = vals[i]; int ix = colB;
            #pragma unroll
            for (int m = 1; m < 16; m <<= 1) {
                float ov = __shfl_xor(v, m, 32);
                int   oi = __shfl_xor(ix, m, 32);
                if (ov > v || (ov == v && oi < ix)) { v = ov; ix = oi; }
            }
            if (r == 0) {
                int row = i + half * 8;           // == batch b (tilesM==1)
                PV[tile * 16 + row] = v;
                PI[tile * 16 + row] = ix;
            }
        }
    }
}

// ------------------- node encoder: H = concat(feat, femb[floor]) @ W + b ----
__global__ void node_encoder_kernel(const float* __restrict__ feat,
                                    const int* __restrict__ floor_id,
                                    const float* __restrict__ femb,
                                    const float* __restrict__ W,
                                    const float* __restrict__ bias,
                                    float* __restrict__ H)
{
    int i = blockIdx.x * blockDim.x + threadIdx.x;
    if (i >= NN * DD) return;
    int n = i >> 6, d = i & 63;
    float s = bias[d];
    #pragma unroll
    for (int k = 0; k < 5; ++k) s += feat[n * 5 + k] * W[k * DD + d];
    int fl = floor_id[n];
    #pragma unroll
    for (int k = 0; k < 8; ++k) s += femb[fl * 8 + k] * W[(5 + k) * DD + d];
    H[i] = s;
}

// ------------------- edge proj stage 1: tmp = relu(edge_attr @ W_e1 + b) ----
__global__ void edge_proj1_kernel(const float* __restrict__ ea,
                                  const float* __restrict__ W,
                                  const float* __restrict__ bias,
                                  float* __restrict__ tmp)
{
    long long i = (long long)blockIdx.x * blockDim.x + threadIdx.x;
    if (i >= (long long)EEC * DD) return;
    long long e = i >> 6; int d = (int)(i & 63);
    float s = bias[d];
    #pragma unroll
    for (int k = 0; k < 5; ++k) s += ea[e * 5 + k] * W[k * DD + d];
    tmp[i] = fmaxf(s, 0.f);
}

// ------------------- GINE scatter: agg[dst] += relu(H[src] + Ee) ------------
__global__ void gine_scatter_kernel(const float* __restrict__ H,
                                    const float* __restrict__ Ee,
                                    const int* __restrict__ src,
                                    const int* __restrict__ dst,
                                    float* __restrict__ agg)
{
    long long i = (long long)blockIdx.x * blockDim.x + threadIdx.x;
    if (i >= (long long)EEC * DD) return;
    int e = (int)(i >> 6), d = (int)(i & 63);
    float m = H[(long long)src[e] * DD + d] + Ee[i];
    m = fmaxf(m, 0.f);
    atomicAdd(&agg[(long long)dst[e] * DD + d], m);
}

// ------------------- traj scatter: agg[td] += w * HR[ts] --------------------
__global__ void traj_scatter_kernel(const float* __restrict__ HR,
                                    const float* __restrict__ w,
                                    const int* __restrict__ ts,
                                    const int* __restrict__ td,
                                    float* __restrict__ agg)
{
    long long i = (long long)blockIdx.x * blockDim.x + threadIdx.x;
    if (i >= (long long)ETT * DD) return;
    int e = (int)(i >> 6), d = (int)(i & 63);
    float m = w[e] * HR[(long long)ts[e] * DD + d];
    atomicAdd(&agg[(long long)td[e] * DD + d], m);
}

// ------------------- row l2 normalize (rows of length 64), 1 wave per row ---
__global__ void l2norm_kernel(const float* __restrict__ X,
                              float* __restrict__ Y, int n)
{
    int wave = threadIdx.x >> 5, lane = threadIdx.x & 31;
    int node = blockIdx.x * (blockDim.x >> 5) + wave;
    if (node >= n) return;
    long long base = (long long)node * DD;
    float v0 = X[base + lane];
    float v1 = X[base + 32 + lane];
    float ss = v0 * v0 + v1 * v1;
    #pragma unroll
    for (int off = 16; off > 0; off >>= 1) ss += __shfl_xor(ss, off, 32);
    float inv = 1.f / fmaxf(sqrtf(ss), 1e-12f);
    Y[base + lane]      = v0 * inv;
    Y[base + 32 + lane] = v1 * inv;
}

// ------------------- pack H_R into B-fragment order for the logits GEMM -----
// PB (as v2f): index p = (tn*16 + s)*32 + lane  ->
//   col = tn*16 + (lane&15), ka = s*4 + (lane>>4)*2,
//   value = { HR[col][ka], HR[col][ka+1] }  (== B[ka][col] of H_R^T)
__global__ void pack_b_kernel(const float* __restrict__ HR,
                              float* __restrict__ PB)
{
    long long i = (long long)blockIdx.x * blockDim.x + threadIdx.x;
    if (i >= (long long)NTILES_N * 16 * 32) return;
    int lane = (int)(i & 31);
    int s    = (int)((i >> 5) & 15);
    int tn   = (int)(i >> 9);
    int col  = tn * 16 + (lane & 15);
    int ka   = s * 4 + (lane >> 4) * 2;
    v2f v;
    v.x = HR[(long long)col * DD + ka];
    v.y = HR[(long long)col * DD + ka + 1];
    ((v2f*)PB)[i] = v;
}

// ------------------- gather encoder inputs ----------------------------------
__global__ void gather_enc_kernel(const float* __restrict__ HR,
                                  const int* __restrict__ pred_seq,
                                  float* __restrict__ enc_inp)
{
    int i = blockIdx.x * blockDim.x + threadIdx.x;
    if (i >= BB * LL * DD) return;
    int bl = i >> 6, d = i & 63;
    int p = pred_seq[bl];
    if (p < 0) p = 0;
    enc_inp[i] = HR[(long long)p * DD + d];
}

// ------------------- bidirectional masked GRU encoder -----------------------
// blockIdx.x = direction (0 fwd, 1 bwd). 256 threads, state in LDS.
// Backward direction reads reversed positions and writes back un-reversed,
// using the involution ridx(t) = (t<len)? len-1-t : t.
__global__ void gru_encoder_kernel(const float* __restrict__ enc_inp,
                                   const int* __restrict__ lengths,
                                   const float* __restrict__ Wih_f, const float* __restrict__ Whh_f,
                                   const float* __restrict__ bih_f, const float* __restrict__ bhh_f,
                                   const float* __restrict__ Wih_b, const float* __restrict__ Whh_b,
                                   const float* __restrict__ bih_b, const float* __restrict__ bhh_b,
                                   float* __restrict__ out_f, float* __restrict__ out_b)
{
    const int dir = blockIdx.x;
    const float* Wih = dir ? Wih_b : Wih_f;
    const float* Whh = dir ? Whh_b : Whh_f;
    const float* bih = dir ? bih_b : bih_f;
    const float* bhh = dir ? bhh_b : bhh_f;
    float* out = dir ? out_b : out_f;

    __shared__ float hs[BB][DD];
    __shared__ float xs[BB][DD];
    int tid = threadIdx.x;
    for (int i = tid; i < BB * DD; i += 256) hs[i >> 6][i & 63] = 0.f;
    __syncthreads();

    for (int t = 0; t < LL; ++t) {
        for (int i = tid; i < BB * DD; i += 256) {
            int b = i >> 6, d = i & 63;
            int len = lengths[b];
            int pos = dir ? ((t < len) ? (len - 1 - t) : t) : t;
            xs[b][d] = enc_inp[((long long)b * LL + pos) * DD + d];
        }
        __syncthreads();
        float hnew[4];
        #pragma unroll
        for (int ii = 0; ii < 4; ++ii) {
            int i = tid + ii * 256;
            int b = i >> 6, d = i & 63;
            float ir = bih[d], iz = bih[64 + d], inn = bih[128 + d];
            float hr = bhh[d], hz = bhh[64 + d], hn = bhh[128 + d];
            for (int k = 0; k < DD; ++k) {
                float xv = xs[b][k], hv = hs[b][k];
                const float* wi = Wih + k * 192 + d;
                const float* wh = Whh + k * 192 + d;
                ir += xv * wi[0]; iz += xv * wi[64]; inn += xv * wi[128];
                hr += hv * wh[0]; hz += hv * wh[64]; hn += hv * wh[128];
            }
            float r  = 1.f / (1.f + expf(-(ir + hr)));
            float zz = 1.f / (1.f + expf(-(iz + hz)));
            float nv = tanhf(inn + r * hn);
            float hcand = (1.f - zz) * nv + zz * hs[b][d];
            int len = lengths[b];
            bool m = (t < len);
            int pos = dir ? (m ? (len - 1 - t) : t) : t;
            out[((long long)b * LL + pos) * DD + d] = m ? hcand : 0.f;
            hnew[ii] = m ? hcand : hs[b][d];
        }
        __syncthreads();
        #pragma unroll
        for (int ii = 0; ii < 4; ++ii) {
            int i = tid + ii * 256;
            hs[i >> 6][i & 63] = hnew[ii];
        }
        __syncthreads();
    }
}

// ------------------- encoder projection + l2norm ----------------------------
__global__ void enc_proj_kernel(const float* __restrict__ out_f,
                                const float* __restrict__ out_b,
                                const float* __restrict__ W_ep,
                                const float* __restrict__ b_ep,
                                float* __restrict__ enc_out)
{
    int row = blockIdx.x;        // b*LL + l
    int d = threadIdx.x;         // 64
    __shared__ float vals[DD];
    __shared__ float nrm;
    float s = b_ep[d];
    const float* f = out_f + (long long)row * DD;
    const float* g = out_b + (long long)row * DD;
    for (int k = 0; k < DD; ++k) s += f[k] * W_ep[k * DD + d];
    for (int k = 0; k < DD; ++k) s += g[k] * W_ep[(DD + k) * DD + d];
    vals[d] = s;
    __syncthreads();
    if (d == 0) {
        float ss = 0.f;
        for (int k = 0; k < DD; ++k) ss += vals[k] * vals[k];
        nrm = 1.f / fmaxf(sqrtf(ss), 1e-12f);
    }
    __syncthreads();
    enc_out[(long long)row * DD + d] = s * nrm;
}

// ------------------- masked mean over time ----------------------------------
__global__ void enc_mean_kernel(const float* __restrict__ enc_out,
                                const int* __restrict__ lengths,
                                float* __restrict__ enc_mean)
{
    int i = blockIdx.x * blockDim.x + threadIdx.x;
    if (i >= BB * DD) return;
    int b = i >> 6, d = i & 63;
    int len = lengths[b];
    float s = 0.f;
    for (int l = 0; l < len; ++l) s += enc_out[((long long)b * LL + l) * DD + d];
    enc_mean[i] = s / (float)(len > 0 ? len : 1);
}

// ------------------- decoder init: h = enc_mean, prev = 0 -------------------
__global__ void dec_init_kernel(const float* __restrict__ enc_mean,
                                float* __restrict__ h_dec,
                                float* __restrict__ prev)
{
    int i = blockIdx.x * blockDim.x + threadIdx.x;
    if (i < BB * DD) { h_dec[i] = enc_mean[i]; prev[i] = 0.f; }
}

// ------------------- one decode step: GRU + attention + z -------------------
// Single block, 256 threads; all state in LDS; h/prev/z live in workspace.
__global__ void dec_step_kernel(const float* __restrict__ enc_out,
                                const int* __restrict__ lengths,
                                const float* __restrict__ Wih, const float* __restrict__ Whh,
                                const float* __restrict__ bih, const float* __restrict__ bhh,
                                const float* __restrict__ W_do, const float* __restrict__ b_do,
                                float* __restrict__ h_g, float* __restrict__ prev_g,
                                float* __restrict__ z_g)
{
    __shared__ float hs[BB][DD], ps[BB][DD], sc[BB][LL], ctx[BB][DD], zt[BB][DD];
    __shared__ float mx[BB], inv[BB], nrm[BB];
    int tid = threadIdx.x;
    for (int i = tid; i < BB * DD; i += 256) {
        hs[i >> 6][i & 63] = h_g[i];
        ps[i >> 6][i & 63] = prev_g[i];
    }
    __syncthreads();
    // --- GRU cell ---
    float hnew[4];
    #pragma unroll
    for (int ii = 0; ii < 4; ++ii) {
        int i = tid + ii * 256;
        int b = i >> 6, d = i & 63;
        float ir = bih[d], iz = bih[64 + d], inn = bih[128 + d];
        float hr = bhh[d], hz = bhh[64 + d], hn = bhh[128 + d];
        for (int k = 0; k < DD; ++k) {
            float xv = ps[b][k], hv = hs[b][k];
            const float* wi = Wih + k * 192 + d;
            const float* wh = Whh + k * 192 + d;
            ir += xv * wi[0]; iz += xv * wi[64]; inn += xv * wi[128];
            hr += hv * wh[0]; hz += hv * wh[64]; hn += hv * wh[128];
        }
        float r  = 1.f / (1.f + expf(-(ir + hr)));
        float zz = 1.f / (1.f + expf(-(iz + hz)));
        float nv = tanhf(inn + r * hn);
        hnew[ii] = (1.f - zz) * nv + zz * hs[b][d];
    }
    __syncthreads();
    #pragma unroll
    for (int ii = 0; ii < 4; ++ii) {
        int i = tid + ii * 256;
        hs[i >> 6][i & 63] = hnew[ii];
        h_g[i] = hnew[ii];
    }
    __syncthreads();
    // --- attention scores ---
    for (int i = tid; i < BB * LL; i += 256) {
        int b = i >> 7, l = i & 127;
        const float* eo = enc_out + ((long long)b * LL + l) * DD;
        float s = 0.f;
        for (int d = 0; d < DD; ++d) s += hs[b][d] * eo[d];
        sc[b][l] = (l < lengths[b]) ? s : -1e9f;
    }
    __syncthreads();
    if (tid < BB) {
        float m = -3.4e38f;
        for (int l = 0; l < LL; ++l) m = fmaxf(m, sc[tid][l]);
        float ssum = 0.f;
        for (int l = 0; l < LL; ++l) ssum += expf(sc[tid][l] - m);
        mx[tid] = m; inv[tid] = 1.f / ssum;
    }
    __syncthreads();
    for (int i = tid; i < BB * LL; i += 256) {
        int b = i >> 7, l = i & 127;
        sc[b][l] = expf(sc[b][l] - mx[b]) * inv[b];
    }
    __syncthreads();
    // --- context ---
    #pragma unroll
    for (int ii = 0; ii < 4; ++ii) {
        int i = tid + ii * 256;
        int b = i >> 6, d = i & 63;
        float s = 0.f;
        for (int l = 0; l < LL; ++l)
            s += sc[b][l] * enc_out[((long long)b * LL + l) * DD + d];
        ctx[b][d] = s;
    }
    __syncthreads();
    // --- z = l2n(concat(h, ctx) @ W_do + b_do) ---
    #pragma unroll
    for (int ii = 0; ii < 4; ++ii) {
        int i = tid + ii * 256;
        int b = i >> 6, d = i & 63;
        float s = b_do[d];
        for (int k = 0; k < DD; ++k) s += hs[b][k] * W_do[k * DD + d];
        for (int k = 0; k < DD; ++k) s += ctx[b][k] * W_do[(DD + k) * DD + d];
        zt[b][d] = s;
    }
    __syncthreads();
    if (tid < BB) {
        float ss = 0.f;
        for (int d = 0; d < DD; ++d) ss += zt[tid][d] * zt[tid][d];
        nrm[tid] = 1.f / fmaxf(sqrtf(ss), 1e-12f);
    }
    __syncthreads();
    #pragma unroll
    for (int ii = 0; ii < 4; ++ii) {
        int i = tid + ii * 256;
        int b = i >> 6, d = i & 63;
        z_g[i] = zt[b][d] * nrm[b];
    }
}

// ------------------- final argmax over per-tile partials, gather next input -
__global__ void argmax_reduce_kernel(const float* __restrict__ PV,
                                     const int* __restrict__ PI,
                                     const float* __restrict__ HR,
                                     float* __restrict__ prev)
{
    int b = blockIdx.x;
    __shared__ float sv[256];
    __shared__ int   si[256];
    int tid = threadIdx.x;
    float best = -3.4e38f; int bi = 0x7fffffff;
    for (int tIdx = tid; tIdx < NTILES_N; tIdx += 256) {
        float v = PV[tIdx * 16 + b];
        int  ix = PI[tIdx * 16 + b];
        if (v > best || (v == best && ix < bi)) { best = v; bi = ix; }
    }
    sv[tid] = best; si[tid] = bi;
    __syncthreads();
    for (int s = 128; s > 0; s >>= 1) {
        if (tid < s) {
            float v = sv[tid + s]; int ix = si[tid + s];
            if (v > sv[tid] || (v == sv[tid] && ix < si[tid])) { sv[tid] = v; si[tid] = ix; }
        }
        __syncthreads();
    }
    int idx = si[0];
    if (tid < DD) prev[b * DD + tid] = HR[(long long)idx * DD + tid];
}

// =====================================================================
// host orchestration
// =====================================================================
extern "C" void kernel_launch(void* const* d_in, const int* in_sizes, int n_in,
                              void* d_out, int out_size, void* d_ws, size_t ws_size,
                              hipStream_t stream)
{
    (void)in_sizes; (void)n_in; (void)out_size; (void)ws_size;
    // ---- inputs (setup_inputs dict order) ----
    const int*   pred_seq  = (const int*)  d_in[0];
    const int*   lengths   = (const int*)  d_in[1];
    const float* node_feat = (const float*)d_in[2];
    const int*   floor_id  = (const int*)  d_in[3];
    const int*   edge_idx  = (const int*)  d_in[4];   // [2,E]
    const float* edge_attr = (const float*)d_in[5];
    const int*   traj_idx  = (const int*)  d_in[6];   // [2,ET]
    const float* traj_w    = (const float*)d_in[7];
    const float* floor_emb = (const float*)d_in[8];
    const float* W_ne = (const float*)d_in[9],  *b_ne = (const float*)d_in[10];
    const float* W_e1 = (const float*)d_in[11], *b_e1 = (const float*)d_in[12];
    const float* W_e2 = (const float*)d_in[13], *b_e2 = (const float*)d_in[14];
    const float* g1_W1 = (const float*)d_in[15], *g1_b1 = (const float*)d_in[16];
    const float* g1_W2 = (const float*)d_in[17], *g1_b2 = (const float*)d_in[18];
    const float* g2_W1 = (const float*)d_in[19], *g2_b1 = (const float*)d_in[20];
    const float* g2_W2 = (const float*)d_in[21], *g2_b2 = (const float*)d_in[22];
    const float* W_t  = (const float*)d_in[23], *b_t  = (const float*)d_in[24];
    const float* Wih_f = (const float*)d_in[25], *Whh_f = (const float*)d_in[26];
    const float* bih_f = (const float*)d_in[27], *bhh_f = (const float*)d_in[28];
    const float* Wih_b = (const float*)d_in[29], *Whh_b = (const float*)d_in[30];
    const float* bih_b = (const float*)d_in[31], *bhh_b = (const float*)d_in[32];
    const float* W_ep = (const float*)d_in[33], *b_ep = (const float*)d_in[34];
    const float* Wih_d = (const float*)d_in[35], *Whh_d = (const float*)d_in[36];
    const float* bih_d = (const float*)d_in[37], *bhh_d = (const float*)d_in[38];
    const float* W_do = (const float*)d_in[39], *b_do = (const float*)d_in[40];

    // ---- workspace layout (floats) ----
    float* ws      = (float*)d_ws;
    float* H       = ws;                              // NN*DD   = 3.2M
    float* agg     = ws + (size_t)NN * DD;            // NN*DD   = 3.2M
    float* Ee      = agg + (size_t)NN * DD;           // EEC*DD  = 12.8M
    float* scratch = Ee + (size_t)EEC * DD;           // 12.8M: tmpE / mid / PB
    float* small_  = scratch + (size_t)EEC * DD;
    float* enc_inp = small_;                          // BB*LL*DD
    float* out_f   = enc_inp + (size_t)BB * LL * DD;
    float* out_b   = out_f   + (size_t)BB * LL * DD;
    float* enc_out = out_b   + (size_t)BB * LL * DD;
    float* enc_mean= enc_out + (size_t)BB * LL * DD;
    float* h_dec   = enc_mean + BB * DD;
    float* prev    = h_dec    + BB * DD;
    float* z       = prev     + BB * DD;
    float* pv      = z        + BB * DD;              // NTILES_N*16 = 50000
    int*   pi      = (int*)(pv + NTILES_N * 16);      // NTILES_N*16 = 50000

    float* logits = (float*)d_out;                               // [BB,LL,NN]
    float* HRout  = logits + (size_t)BB * LL * NN;               // [NN,DD]

    const int* e_src = edge_idx;
    const int* e_dst = edge_idx + EEC;
    const int* t_src = traj_idx;
    const int* t_dst = traj_idx + ETT;

    // ---- 1. node encoder ----
    node_encoder_kernel<<<(NN * DD) / 256, 256, 0, stream>>>(
        node_feat, floor_id, floor_emb, W_ne, b_ne, H);

    // ---- 2. edge MLP: Ee = relu(ea@W_e1+b) @ W_e2 + b ----
    float* tmpE = scratch;
    edge_proj1_kernel<<<(int)(((long long)EEC * DD) / 256), 256, 0, stream>>>(
        edge_attr, W_e1, b_e1, tmpE);
    {
        int blocks = ((EEC / 16) * (DD / 16) + 3) / 4;
        gemm_wmma_kernel<DD, DD, false, false, false><<<blocks, 128, 0, stream>>>(
            tmpE, nullptr, W_e2, b_e2, Ee, EEC, DD, 1.0f, 0, nullptr, nullptr);
    }

    // ---- 3. two GINE layers ----
    float* mid = scratch;   // tmpE dead now; mid = NN*MLPH = 6.4M
    const float* Ws1[2] = {g1_W1, g2_W1}; const float* bs1[2] = {g1_b1, g2_b1};
    const float* Ws2[2] = {g1_W2, g2_W2}; const float* bs2[2] = {g1_b2, g2_b2};
    for (int layer = 0; layer < 2; ++layer) {
        hipMemsetAsync(agg, 0, (size_t)NN * DD * sizeof(float), stream);
        gine_scatter_kernel<<<(int)(((long long)EEC * DD) / 256), 256, 0, stream>>>(
            H, Ee, e_src, e_dst, agg);
        {
            int blocks = ((NN / 16) * (MLPH / 16) + 3) / 4;
            gemm_wmma_kernel<DD, MLPH, true, false, false><<<blocks, 128, 0, stream>>>(
                H, agg, Ws1[layer], bs1[layer], mid, NN, MLPH, 1.0f, 1, nullptr, nullptr);
        }
        {
            int blocks = ((NN / 16) * (DD / 16) + 3) / 4;
            gemm_wmma_kernel<MLPH, DD, false, false, false><<<blocks, 128, 0, stream>>>(
                mid, nullptr, Ws2[layer], bs2[layer], H, NN, DD, 1.0f, 1, nullptr, nullptr);
        }
    }
    l2norm_kernel<<<NN / 8, 256, 0, stream>>>(H, HRout, NN);

    // ---- 4. trajectory weighted GCN ----
    hipMemsetAsync(agg, 0, (size_t)NN * DD * sizeof(float), stream);
    traj_scatter_kernel<<<(int)(((long long)ETT * DD) / 256), 256, 0, stream>>>(
        HRout, traj_w, t_src, t_dst, agg);
    {
        int blocks = ((NN / 16) * (DD / 16) + 3) / 4;
        gemm_wmma_kernel<DD, DD, true, false, false><<<blocks, 128, 0, stream>>>(
            HRout, agg, W_t, b_t, H, NN, DD, 1.0f, 1, nullptr, nullptr);
    }
    l2norm_kernel<<<NN / 8, 256, 0, stream>>>(H, HRout, NN);

    // ---- 5. pack H_R^T into B-fragment order once for the decode loop ----
    float* PB = scratch;    // NTILES_N*16*32*2 = 3.2M floats; mid dead
    pack_b_kernel<<<(NTILES_N * 16 * 32) / 256, 256, 0, stream>>>(HRout, PB);

    // ---- 6. encoder ----
    gather_enc_kernel<<<(BB * LL * DD) / 256, 256, 0, stream>>>(HRout, pred_seq, enc_inp);
    gru_encoder_kernel<<<2, 256, 0, stream>>>(enc_inp, lengths,
        Wih_f, Whh_f, bih_f, bhh_f, Wih_b, Whh_b, bih_b, bhh_b, out_f, out_b);
    enc_proj_kernel<<<BB * LL, 64, 0, stream>>>(out_f, out_b, W_ep, b_ep, enc_out);
    enc_mean_kernel<<<(BB * DD + 255) / 256, 256, 0, stream>>>(enc_out, lengths, enc_mean);

    // ---- 7. greedy decode loop ----
    dec_init_kernel<<<(BB * DD + 255) / 256, 256, 0, stream>>>(enc_mean, h_dec, prev);
    const int logit_blocks = (NTILES_N + 3) / 4;
    for (int t = 0; t < LL; ++t) {
        dec_step_kernel<<<1, 256, 0, stream>>>(enc_out, lengths,
            Wih_d, Whh_d, bih_d, bhh_d, W_do, b_do, h_dec, prev, z);
        // logits[:, t, :] = TEMP * z @ H_R^T  (packed B, fused partial argmax)
        gemm_wmma_kernel<DD, NN, false, true, true><<<logit_blocks, 128, 0, stream>>>(
            z, nullptr, PB, nullptr, logits + (size_t)t * NN,
            BB, (long long)LL * NN, TEMP, 0, pv, pi);
        argmax_reduce_kernel<<<BB, 256, 0, stream>>>(pv, pi, HRout, prev);
    }
}